// ECEWithProbabilities_21423296872466
// MI455X (gfx1250) — compile-verified
//
#include <hip/hip_runtime.h>
#include <hip/hip_bf16.h>

#define NUM_BINS 15
#define HIST_PAD 48   // 3 fields * 16 bins (bin 15 unused, stays 0)

typedef __attribute__((ext_vector_type(2))) float v2f;
typedef __attribute__((ext_vector_type(8))) float v8f;

// ---------------------------------------------------------------------------
// Kernel 1: gather + per-block 15-bin histogram in LDS (ds_add_f32),
// write 48-float partial per block to workspace.
// ---------------------------------------------------------------------------
__global__ void ece_hist_kernel(const float* __restrict__ prob,
                                const int*   __restrict__ target_pred,
                                float*       __restrict__ partials,
                                int N, int C)
{
    __shared__ float h[HIST_PAD];
    const int tid = threadIdx.x;
    if (tid < HIST_PAD) h[tid] = 0.0f;
    __syncthreads();

    const float step = 1.0f / 15.0f;          // jnp.linspace delta in f32
    const int stride = gridDim.x * blockDim.x;
    const int2* __restrict__ tp2 = (const int2*)target_pred;

    for (int i = blockIdx.x * blockDim.x + tid; i < N; i += stride) {
        // prefetch the next strided (label,pred) pair -> global_prefetch_b8
        if (i + stride < N) __builtin_prefetch(&tp2[i + stride], 0, 0);

        int2 lp = tp2[i];                      // x = label, y = pred (coalesced 8B)
        const int   label = lp.x;
        const int   pred  = lp.y;
        const float conf  = prob[(size_t)i * (size_t)C + (size_t)pred]; // gather

        // bin = count(edges < conf) - 1 ; edges[j] = j * (1/15) in f32
        int cnt = 0;
#pragma unroll
        for (int j = 0; j <= NUM_BINS; ++j)
            cnt += ((float)j * step < conf) ? 1 : 0;
        int bin = cnt - 1;

        if (bin >= 0 && bin < NUM_BINS) {      // valid mask (w == 1)
            const float acc = (pred == label) ? 1.0f : 0.0f;
            // relaxed WG-scope fp adds on LDS -> ds_add_f32 (no CAS loop)
            __hip_atomic_fetch_add(&h[ 0 + bin], 1.0f, __ATOMIC_RELAXED, __HIP_MEMORY_SCOPE_WORKGROUP);
            __hip_atomic_fetch_add(&h[16 + bin], conf, __ATOMIC_RELAXED, __HIP_MEMORY_SCOPE_WORKGROUP);
            __hip_atomic_fetch_add(&h[32 + bin], acc,  __ATOMIC_RELAXED, __HIP_MEMORY_SCOPE_WORKGROUP);
        }
    }
    __syncthreads();
    if (tid < HIST_PAD) partials[blockIdx.x * HIST_PAD + tid] = h[tid];
}

// ---------------------------------------------------------------------------
// Kernel 2: single wave32. Sum NB partials with v_wmma_f32_16x16x4_f32:
//   D(16x16) += ones(16x4) x B(4x16)
// B column n = bin n of one field; 4 K-rows = 4 block-partials.
// A == all-ones makes the result invariant to A/B register-layout details:
// every element of column n of D equals sum over all blocks of bin n.
// ---------------------------------------------------------------------------
__global__ void ece_reduce_kernel(const float* __restrict__ partials,
                                  int NB,                    // multiple of 4
                                  float* __restrict__ out,
                                  int N)
{
    const int lane = threadIdx.x;       // 0..31, one full wave (EXEC all ones)
    const int col  = lane & 15;         // bin column
    const int half = lane >> 4;         // 0: K rows {0,1}; 1: K rows {2,3}

    v2f ones; ones.x = 1.0f; ones.y = 1.0f;
    v8f acc[3];
#pragma unroll
    for (int f = 0; f < 3; ++f) acc[f] = (v8f){0,0,0,0,0,0,0,0};

    for (int b = 0; b < NB; b += 4) {
        const int r0 = b + half * 2;    // this lane-half's first K row (block)
        const int r1 = r0 + 1;
#pragma unroll
        for (int f = 0; f < 3; ++f) {
            v2f bv;
            bv.x = partials[r0 * HIST_PAD + f * 16 + col];
            bv.y = partials[r1 * HIST_PAD + f * 16 + col];
            // 8 args: (neg_a, A, neg_b, B, c_mod, C, reuse_a, reuse_b)
            acc[f] = __builtin_amdgcn_wmma_f32_16x16x4_f32(
                false, ones, false, bv, (short)0, acc[f], false, false);
        }
    }

    // Every row of D holds the column sums; element 0 of the v8f at lane l
    // is the sum for bin (l & 15).
    const float cnt = acc[0][0];
    const float sc  = acc[1][0];
    const float sa  = acc[2][0];

    __shared__ float terms[16];
    float t = 0.0f;
    if (cnt > 0.0f)
        t = (cnt / (float)N) * fabsf(sa / cnt - sc / cnt);
    if (lane < NUM_BINS) terms[lane] = t;
    __syncthreads();

    if (lane == 0) {
        float ece = 0.0f;
#pragma unroll
        for (int j = 0; j < NUM_BINS; ++j) ece += terms[j];
        out[0] = ece;
    }
}

// ---------------------------------------------------------------------------
// Launcher
// ---------------------------------------------------------------------------
extern "C" void kernel_launch(void* const* d_in, const int* in_sizes, int n_in,
                              void* d_out, int out_size, void* d_ws, size_t ws_size,
                              hipStream_t stream)
{
    const float* prob = (const float*)d_in[0];
    const int*   tp   = (const int*)d_in[1];

    const int N = in_sizes[1] / 2;              // 1,000,000
    const int C = (N > 0) ? (in_sizes[0] / N) : 1;   // 100

    float* partials = (float*)d_ws;

    // 1024 blocks of 256 threads: ~262K threads for 1M latency-bound gathers.
    int NB = 1024;
    size_t need = (size_t)NB * HIST_PAD * sizeof(float);
    if (ws_size < need) {
        int m = (int)(ws_size / (HIST_PAD * sizeof(float)));
        NB = m & ~3;                            // keep multiple of 4 for WMMA K
        if (NB < 4) NB = 4;
    }

    ece_hist_kernel<<<NB, 256, 0, stream>>>(prob, tp, partials, N, C);
    ece_reduce_kernel<<<1, 32, 0, stream>>>(partials, NB, (float*)d_out, N);
}